// HungarianMatcher_33569464386067
// MI455X (gfx1250) — compile-verified
//
#include <hip/hip_runtime.h>
#include <hip/hip_bf16.h>
#include <cstddef>

// HungarianMatcher cost matrix for MI455X (gfx1250, wave32).
//
// dist2 is computed as a rank-4 GEMM via V_WMMA_F32_16X16X4_F32:
//   A[q,:] = [x_q, y_q, x_q^2+y_q^2, 1]
//   B[:,t] = [-2*x_t, -2*y_t, 1, x_t^2+y_t^2]
// One wave -> one 16x16 tile of dist2 per wmma, then sqrt/exp2/mask-gather
// epilogue. Memory-bound: ~268MB mask stream + 67MB output => ~14us floor.
// All global gathers/stores use 32-bit offsets from uniform bases so the
// compiler emits saddr+voffset addressing (no per-lane 64-bit address math).

typedef __attribute__((ext_vector_type(2))) float v2f;
typedef __attribute__((ext_vector_type(8))) float v8f;

// exp(-d2/200) == exp2(d2 * (-log2(e)/200))
#define NEG_LOG2E_OVER_2SIG2 (-0.007213475204444817f)

// ---------------------------------------------------------------------------
// Prep: per-query and per-target coefficient vectors into workspace.
// ---------------------------------------------------------------------------
__global__ __launch_bounds__(256) void hm_prep(
    const float* __restrict__ logits,   // [Q,2]
    const float* __restrict__ points,   // [Q,2] normalized
    const float* __restrict__ tpts,     // [T,2] absolute
    const int* __restrict__ ph, const int* __restrict__ pw,
    float* __restrict__ qx, float* __restrict__ qy, float* __restrict__ qs,
    float* __restrict__ np0, float* __restrict__ np1, int* __restrict__ moff,
    float* __restrict__ tb0, float* __restrict__ tb1, float* __restrict__ tcs,
    int Q, int T)
{
    const int i = blockIdx.x * blockDim.x + threadIdx.x;
    const int H = ph[0], W = pw[0];
    if (i < Q) {
        // absolute coords: dim0 scaled by H, dim1 by W (reference convention)
        float x = points[2 * i + 0] * (float)H;
        float y = points[2 * i + 1] * (float)W;
        qx[i] = x;
        qy[i] = y;
        qs[i] = x * x + y * y;
        // 2-class softmax -> negated probabilities
        float l0 = logits[2 * i + 0], l1 = logits[2 * i + 1];
        float mx = fmaxf(l0, l1);
        float e0 = __expf(l0 - mx), e1 = __expf(l1 - mx);
        float inv = 1.0f / (e0 + e1);
        np0[i] = -e0 * inv;
        np1[i] = -e1 * inv;
        // mask lookup offset: xs from dim0 (col index), ys from dim1 (row index)
        int xs = (int)x; xs = xs < 0 ? 0 : (xs > H - 1 ? H - 1 : xs);
        int ys = (int)y; ys = ys < 0 ? 0 : (ys > W - 1 ? W - 1 : ys);
        moff[i] = ys * W + xs;
    }
    if (i < T) {
        float tx = tpts[2 * i + 0], ty = tpts[2 * i + 1];
        tb0[i] = -2.0f * tx;
        tb1[i] = -2.0f * ty;
        tcs[i] = tx * tx + ty * ty;
    }
}

// ---------------------------------------------------------------------------
// Main: one wave per 16-row tile x tilesPerGroup 16-col tiles.
// colGroup is the slow wave index so concurrently resident waves share the
// same ~32MB window of mask slices (fits easily in 192MB L2).
// ---------------------------------------------------------------------------
__global__ __launch_bounds__(256) void hm_cost(
    const float* __restrict__ qx, const float* __restrict__ qy,
    const float* __restrict__ qs,
    const float* __restrict__ np0, const float* __restrict__ np1,
    const int* __restrict__ moff,
    const float* __restrict__ tb0, const float* __restrict__ tb1,
    const float* __restrict__ tcs,
    const int* __restrict__ tids,
    const unsigned char* __restrict__ masks,   // [T,H,W] bool bytes
    const int* __restrict__ ph, const int* __restrict__ pw,
    float* __restrict__ out,                   // [Q,T]
    int rowTiles, int colGroups, int tilesPerGroup, int T)
{
    const int lane     = threadIdx.x & 31;
    const int waveId   = blockIdx.x * (blockDim.x >> 5) + (threadIdx.x >> 5);
    const int colGroup = waveId / rowTiles;     // slow index -> L2 locality on masks
    const int rowTile  = waveId % rowTiles;
    if (colGroup >= colGroups) return;          // wave-uniform exit (EXEC stays all-1s)

    const int      qBase = rowTile << 4;
    const unsigned hw    = (unsigned)(ph[0] * pw[0]);
    const int      half  = lane >> 4;           // 0: lanes 0-15, 1: lanes 16-31
    const int      l16   = lane & 15;

    // A operand (16x4 f32): lanes 0-15 hold K=0,1 ; lanes 16-31 hold K=2,3.
    // K columns: [x, y, x^2+y^2, 1]
    v2f a;
    {
        const int m = qBase + l16;              // M = lane % 16
        if (half == 0) { a.x = qx[m]; a.y = qy[m]; }
        else           { a.x = qs[m]; a.y = 1.0f; }
    }

    // Per-row scalars for this lane's 8 output rows (M = qBase + r + 8*half),
    // plus 32-bit output base offsets (out is [Q,T], < 2^31 elements).
    float    rn0[8], rn1[8];
    unsigned rmo[8], obase[8];
#pragma unroll
    for (int r = 0; r < 8; ++r) {
        const int M = qBase + r + half * 8;
        rn0[r]   = np0[M];
        rn1[r]   = np1[M];
        rmo[r]   = (unsigned)moff[M];
        obase[r] = (unsigned)M * (unsigned)T + (unsigned)l16;
    }

    const int tcol0 = (colGroup * tilesPerGroup) << 4;
    for (int ct = 0; ct < tilesPerGroup; ++ct) {
        const int tBase = tcol0 + (ct << 4);
        const int tcol  = tBase + l16;          // N = lane % 16 for both halves

        // B operand (4x16 f32): lanes 0-15 hold K=0,1 ; lanes 16-31 hold K=2,3.
        // K rows: [-2*tx, -2*ty, 1, tx^2+ty^2]
        v2f b;
        if (half == 0) { b.x = tb0[tcol]; b.y = tb1[tcol]; }
        else           { b.x = 1.0f;      b.y = tcs[tcol]; }

        v8f c = {};
        // (neg_a, A, neg_b, B, c_mod, C, reuse_a, reuse_b)
        v8f d = __builtin_amdgcn_wmma_f32_16x16x4_f32(
            false, a, false, b, (short)0, c, false, false);

        const int      tid      = tids[tcol];
        // 32-bit offset into the uniform mask base (< 2^31 here) so the
        // compiler can use saddr + zext(vgpr32) global addressing.
        const unsigned sliceOff = (unsigned)tcol * hw;

#pragma unroll
        for (int r = 0; r < 8; ++r) {
            const float d2 = d[r];
            const float cp = __builtin_amdgcn_sqrtf(d2);                    // cost_point
            const float g  = __builtin_amdgcn_exp2f(d2 * NEG_LOG2E_OVER_2SIG2);
            const unsigned char ins = masks[(size_t)(sliceOff + rmo[r])];
            const float cm = ins ? (1.0f - g) : 10.0f;                      // cost_mask
            const float cc = tid ? rn1[r] : rn0[r];                         // cost_class
            out[(size_t)(obase[r] + (unsigned)tBase)] = cp + cc + cm;
        }
    }
}

// ---------------------------------------------------------------------------
extern "C" void kernel_launch(void* const* d_in, const int* in_sizes, int n_in,
                              void* d_out, int out_size, void* d_ws, size_t ws_size,
                              hipStream_t stream) {
    const float*         logits = (const float*)d_in[0];          // [bs,nq,2]
    const float*         points = (const float*)d_in[1];          // [bs,nq,2]
    const float*         tpts   = (const float*)d_in[2];          // [T,2]
    const int*           tids   = (const int*)d_in[3];            // [T]
    const unsigned char* masks  = (const unsigned char*)d_in[4];  // [T,H,W] bool
    const int*           ph     = (const int*)d_in[5];            // img_h
    const int*           pw     = (const int*)d_in[6];            // img_w

    const int Q = in_sizes[1] / 2;   // 16384
    const int T = in_sizes[3];       // 1024

    // Workspace layout (~405 KB)
    float* qx  = (float*)d_ws;
    float* qy  = qx + Q;
    float* qs  = qy + Q;
    float* np0 = qs + Q;
    float* np1 = np0 + Q;
    int*   mof = (int*)(np1 + Q);
    float* tb0 = (float*)(mof + Q);
    float* tb1 = tb0 + T;
    float* tcs = tb1 + T;

    const int prepN = (Q > T) ? Q : T;
    hm_prep<<<(prepN + 255) / 256, 256, 0, stream>>>(
        logits, points, tpts, ph, pw,
        qx, qy, qs, np0, np1, mof, tb0, tb1, tcs, Q, T);

    const int rowTiles = Q / 16;                   // 1024
    const int colTiles = T / 16;                   // 64
    int colGroups = 8;
    if (colTiles % colGroups) colGroups = 1;       // safety fallback
    const int tilesPerGroup = colTiles / colGroups;
    const int totalWaves = rowTiles * colGroups;   // 8192 waves
    const int wavesPerBlock = 256 / 32;
    const int blocks = (totalWaves + wavesPerBlock - 1) / wavesPerBlock;

    hm_cost<<<blocks, 256, 0, stream>>>(
        qx, qy, qs, np0, np1, mof, tb0, tb1, tcs,
        tids, masks, ph, pw, (float*)d_out,
        rowTiles, colGroups, tilesPerGroup, T);
}